// GCNModel_52656299049200
// MI455X (gfx1250) — compile-verified
//
#include <hip/hip_runtime.h>

#define N_NODES 20000
#define N_EDGES 1280000
#define CH      256
#define K_TOT   512        // [agg*invdeg | x] concatenated K
#define LDSTR   520        // LDS row stride in floats (even; 520*4 % 16 == 0 for b128)

typedef __attribute__((ext_vector_type(2))) float v2f;
typedef __attribute__((ext_vector_type(8))) float v8f;

// ---------------------------------------------------------------- utilities
__global__ void zero_kernel(float4* p, int n4) {
    int i = blockIdx.x * blockDim.x + threadIdx.x;
    if (i < n4) p[i] = make_float4(0.f, 0.f, 0.f, 0.f);
}

__global__ void init_out_kernel(float* out, const float* fcb) {
    out[0] = fcb[0];
}

// ---------------------------------------------------------------- degree
__global__ void deg_kernel(const long long* __restrict__ ei, float* __restrict__ deg) {
    int e = blockIdx.x * blockDim.x + threadIdx.x;
    if (e < N_EDGES) {
        int dst = (int)ei[N_EDGES + e];
        atomicAdd(&deg[dst], 1.0f);
    }
}

__global__ void invdeg_kernel(float* __restrict__ deg) {
    int n = blockIdx.x * blockDim.x + threadIdx.x;
    if (n < N_NODES) {
        deg[n] = 1.0f / fmaxf(deg[n], 1.0f);
    }
}

// ---------------------------------------------------------------- edge aggregation
// 4 edges per 256-thread block, 64 threads per edge: one global_load_b128
// gather + 4 global_atomic_add_f32 per thread. Feature matrices (20.5 MB)
// are L2-resident (192 MB L2), so this is L2-bandwidth work, not HBM.
__global__ __launch_bounds__(256) void aggregate_kernel(const long long* __restrict__ ei,
                                                        const float* __restrict__ feat,
                                                        float* __restrict__ agg) {
    const int eslot = threadIdx.x >> 6;          // 0..3
    const int t     = threadIdx.x & 63;          // float4 slot within the row
    const long long e = (long long)blockIdx.x * 4 + eslot;
    if (e >= N_EDGES) return;                    // uniform per 64-lane group
    const int src = (int)ei[e];
    const int dst = (int)ei[N_EDGES + e];
    const float4 v = ((const float4*)(feat + (size_t)src * CH))[t];
    float* d = agg + (size_t)dst * CH + (size_t)t * 4;
    atomicAdd(d + 0, v.x);
    atomicAdd(d + 1, v.y);
    atomicAdd(d + 2, v.z);
    atomicAdd(d + 3, v.w);
}

// ---------------------------------------------------------------- layer-1 fused GEMM
// h1 = relu([agg*invdeg | x] @ [W1_l | W1_r]^T + b1), 20000x512 @ 512x256.
// Block = 4 waves = one 16-row tile; wave w owns cols [64w, 64w+64).
// A tile staged via GLOBAL_LOAD_ASYNC_TO_LDS_B128 (ASYNCcnt path); the
// per-row inv_deg scale is applied to the A fragment inside the K loop,
// which is split at k=256 (wave-uniform) so each half reads one weight
// matrix with no per-iteration select. 512 v_wmma_f32_16x16x4_f32 per wave.
__global__ __launch_bounds__(128) void sage1_wmma_kernel(const float* __restrict__ agg,
                                                         const float* __restrict__ x,
                                                         const float* __restrict__ invdeg,
                                                         const float* __restrict__ W1l,
                                                         const float* __restrict__ W1r,
                                                         const float* __restrict__ bias1,
                                                         float* __restrict__ h1) {
    __shared__ float Atile[16 * LDSTR];

    const int tid     = threadIdx.x;
    const int rowBase = blockIdx.x * 16;

    // Generic pointers to LDS keep the byte offset in the low 32 bits
    // (ISA 10.2: flat->LDS truncates to addr[31:0]).
    const unsigned ldsBase = (unsigned)(size_t)&Atile[0];

    // Async-stage the raw agg rows into cols [0,256) and x rows into
    // cols [256,512): 2 x 1024 b128 chunks, 8 chunks per thread per half,
    // consecutive lanes -> consecutive 16B of one row (coalesced).
    for (int c = tid; c < 16 * 64; c += 128) {
        const int r = c >> 6, f4 = c & 63;
        const unsigned lds = ldsBase + (unsigned)((r * LDSTR + f4 * 4) * 4);
        const unsigned long long g =
            (unsigned long long)(size_t)(agg + (size_t)(rowBase + r) * CH + (size_t)f4 * 4);
        asm volatile("global_load_async_to_lds_b128 %0, %1, off"
                     :: "v"(lds), "v"(g) : "memory");
    }
    for (int c = tid; c < 16 * 64; c += 128) {
        const int r = c >> 6, f4 = c & 63;
        const unsigned lds = ldsBase + (unsigned)((r * LDSTR + CH + f4 * 4) * 4);
        const unsigned long long g =
            (unsigned long long)(size_t)(x + (size_t)(rowBase + r) * CH + (size_t)f4 * 4);
        asm volatile("global_load_async_to_lds_b128 %0, %1, off"
                     :: "v"(lds), "v"(g) : "memory");
    }
    asm volatile("s_wait_asynccnt 0" ::: "memory");
    __syncthreads();

    const int wave  = tid >> 5;
    const int lane  = tid & 31;
    const int row   = lane & 15;          // A-matrix M for this lane
    const int khalf = (lane >> 4) << 1;   // lanes 0-15: K=+0/+1, lanes 16-31: K=+2/+3
    const int nBase = wave * 64;
    const int nLane = lane & 15;          // B/D column within a 16-wide tile
    const int n0    = nBase + nLane;

    const float vinv = invdeg[rowBase + row];   // per-row mean scale

    v8f acc0 = {}, acc1 = {}, acc2 = {}, acc3 = {};

    // --- K half 1: mean = agg*invdeg against W1_l ---
    for (int kk = 0; kk < CH; kk += 4) {
        const int ka = kk + khalf;
        v2f a = *(const v2f*)&Atile[row * LDSTR + ka];
        a = a * vinv;
        v2f b0 = *(const v2f*)&W1l[(size_t)(n0     ) * CH + ka];
        v2f b1 = *(const v2f*)&W1l[(size_t)(n0 + 16) * CH + ka];
        v2f b2 = *(const v2f*)&W1l[(size_t)(n0 + 32) * CH + ka];
        v2f b3 = *(const v2f*)&W1l[(size_t)(n0 + 48) * CH + ka];
        acc0 = __builtin_amdgcn_wmma_f32_16x16x4_f32(false, a, false, b0, (short)0, acc0, false, false);
        acc1 = __builtin_amdgcn_wmma_f32_16x16x4_f32(false, a, false, b1, (short)0, acc1, false, false);
        acc2 = __builtin_amdgcn_wmma_f32_16x16x4_f32(false, a, false, b2, (short)0, acc2, false, false);
        acc3 = __builtin_amdgcn_wmma_f32_16x16x4_f32(false, a, false, b3, (short)0, acc3, false, false);
    }
    // --- K half 2: x against W1_r ---
    for (int kk = 0; kk < CH; kk += 4) {
        const int ka = kk + khalf;
        v2f a = *(const v2f*)&Atile[row * LDSTR + CH + ka];
        v2f b0 = *(const v2f*)&W1r[(size_t)(n0     ) * CH + ka];
        v2f b1 = *(const v2f*)&W1r[(size_t)(n0 + 16) * CH + ka];
        v2f b2 = *(const v2f*)&W1r[(size_t)(n0 + 32) * CH + ka];
        v2f b3 = *(const v2f*)&W1r[(size_t)(n0 + 48) * CH + ka];
        acc0 = __builtin_amdgcn_wmma_f32_16x16x4_f32(false, a, false, b0, (short)0, acc0, false, false);
        acc1 = __builtin_amdgcn_wmma_f32_16x16x4_f32(false, a, false, b1, (short)0, acc1, false, false);
        acc2 = __builtin_amdgcn_wmma_f32_16x16x4_f32(false, a, false, b2, (short)0, acc2, false, false);
        acc3 = __builtin_amdgcn_wmma_f32_16x16x4_f32(false, a, false, b3, (short)0, acc3, false, false);
    }

    // Epilogue: D layout -> VGPR v holds (M = v [+8 for upper lanes], N = lane&15).
    const int mBase = rowBase + ((lane >> 4) << 3);
    v8f accs[4] = {acc0, acc1, acc2, acc3};
    #pragma unroll
    for (int t = 0; t < 4; ++t) {
        const int   col  = nBase + t * 16 + nLane;
        const float bval = bias1[col];
        #pragma unroll
        for (int v = 0; v < 8; ++v) {
            float val = accs[t][v] + bval;
            h1[(size_t)(mBase + v) * CH + col] = fmaxf(val, 0.0f);
        }
    }
}

// ---------------------------------------------------------------- layer-2 + readout
// One wave per node: two 256-length dots, shuffle-reduce, relu, scale by fc_w,
// atomic accumulate into scalar output.
__global__ __launch_bounds__(256) void final_kernel(const float* __restrict__ agg2,
                                                    const float* __restrict__ h1,
                                                    const float* __restrict__ invdeg,
                                                    const float* __restrict__ W2l,
                                                    const float* __restrict__ b2,
                                                    const float* __restrict__ W2r,
                                                    const float* __restrict__ fcw,
                                                    float* __restrict__ out) {
    const int wave = threadIdx.x >> 5;
    const int lane = threadIdx.x & 31;
    const int node = blockIdx.x * 8 + wave;
    if (node >= N_NODES) return;

    float s1 = 0.f, s2 = 0.f;
    for (int i = lane; i < CH; i += 32) {
        s1 += agg2[(size_t)node * CH + i] * W2l[i];
        s2 += h1  [(size_t)node * CH + i] * W2r[i];
    }
    #pragma unroll
    for (int off = 16; off > 0; off >>= 1) {
        s1 += __shfl_xor(s1, off, 32);
        s2 += __shfl_xor(s2, off, 32);
    }
    if (lane == 0) {
        float h2 = fmaxf(invdeg[node] * s1 + b2[0] + s2, 0.0f);
        atomicAdd(out, h2 * fcw[node]);
    }
}

// ---------------------------------------------------------------- launch
extern "C" void kernel_launch(void* const* d_in, const int* in_sizes, int n_in,
                              void* d_out, int out_size, void* d_ws, size_t ws_size,
                              hipStream_t stream) {
    const float*     x   = (const float*)d_in[0];
    const long long* ei  = (const long long*)d_in[1];  // [2, N_EDGES] int64
    const float*     W1l = (const float*)d_in[2];
    const float*     b1  = (const float*)d_in[3];
    const float*     W1r = (const float*)d_in[4];
    const float*     W2l = (const float*)d_in[5];
    const float*     b2  = (const float*)d_in[6];
    const float*     W2r = (const float*)d_in[7];
    const float*     fcw = (const float*)d_in[8];
    const float*     fcb = (const float*)d_in[9];
    float*           out = (float*)d_out;

    // Workspace layout (floats): invdeg | agg | agg2 | h1   (~61.5 MB total)
    float* ws     = (float*)d_ws;
    float* invdeg = ws;                                  // 20000
    float* agg    = invdeg + N_NODES;                    // 5,120,000
    float* agg2   = agg + (size_t)N_NODES * CH;          // 5,120,000
    float* h1     = agg2 + (size_t)N_NODES * CH;         // 5,120,000

    // Zero invdeg+agg+agg2 (contiguous region, count divisible by 4).
    const int zeroN4 = (N_NODES + 2 * N_NODES * CH) / 4; // 2,565,000 float4
    zero_kernel<<<(zeroN4 + 255) / 256, 256, 0, stream>>>((float4*)ws, zeroN4);

    deg_kernel<<<(N_EDGES + 255) / 256, 256, 0, stream>>>(ei, invdeg);
    invdeg_kernel<<<(N_NODES + 255) / 256, 256, 0, stream>>>(invdeg);

    aggregate_kernel<<<N_EDGES / 4, 256, 0, stream>>>(ei, x, agg);

    sage1_wmma_kernel<<<N_NODES / 16, 128, 0, stream>>>(agg, x, invdeg, W1l, W1r, b1, h1);

    aggregate_kernel<<<N_EDGES / 4, 256, 0, stream>>>(ei, h1, agg2);

    init_out_kernel<<<1, 1, 0, stream>>>(out, fcb);
    final_kernel<<<N_NODES / 8, 256, 0, stream>>>(agg2, h1, invdeg, W2l, b2, W2r, fcw, out);
}